// MeanPoolNetwork_87720412054264
// MI455X (gfx1250) — compile-verified
//
#include <hip/hip_runtime.h>
#include <hip/hip_bf16.h>

#define N_NODES 100000
#define N_EDGES 1600000
#define F_IN 128
#define N_GRAPHS 1000
#define NUM_CLASSES 2

typedef __attribute__((ext_vector_type(2))) float v2f;
typedef __attribute__((ext_vector_type(8))) float v8f;

// ---------------------------------------------------------------- utilities
__global__ void fill_kernel(float* __restrict__ p, float v, long n) {
    long i = (long)blockIdx.x * blockDim.x + threadIdx.x;
    if (i < n) p[i] = v;
}

// deg[dst] += w  (self-loop "1" is the fill initial value)
__global__ void deg_kernel(const int* __restrict__ dst,
                           const float* __restrict__ ew,
                           float* __restrict__ deg, int nE) {
    int e = blockIdx.x * blockDim.x + threadIdx.x;
    if (e < nE) atomicAdd(deg + dst[e], ew[e]);
}

__global__ void dinv_kernel(float* __restrict__ deg, int n) {
    int i = blockIdx.x * blockDim.x + threadIdx.x;
    if (i < n) {
        float d = deg[i];
        deg[i] = (d > 0.0f) ? rsqrtf(fmaxf(d, 1e-12f)) : 0.0f;
    }
}

__global__ void normw_kernel(const int* __restrict__ src, const int* __restrict__ dst,
                             const float* __restrict__ ew,
                             const float* __restrict__ dinv,
                             float* __restrict__ nw, int nE) {
    int e = blockIdx.x * blockDim.x + threadIdx.x;
    if (e < nE) nw[e] = dinv[src[e]] * ew[e] * dinv[dst[e]];
}

// ------------------------------------------------- f32 WMMA GEMM (16x32 tiles)
// C[M,N] = A[M,K] @ B[K,N]  (+bias, +relu optional). One wave computes a 16x32
// output tile via two independent V_WMMA_F32_16X16X4_F32 accumulator chains
// (doubles WMMA ILP, halves A-fragment loads). N must be a multiple of 32,
// K a multiple of 4. OOB M rows are clamped to row 0 (their garbage C rows are
// never stored), so the inner loop has no predication -> EXEC stays all-ones.
__global__ __launch_bounds__(256)
void gemm16_wmma(const float* __restrict__ A, const float* __restrict__ B,
                 const float* __restrict__ bias, float* __restrict__ C,
                 int M, int N, int K, int do_relu) {
    const int lane = threadIdx.x & 31;
    const int wavesPerBlock = blockDim.x >> 5;
    const int wave = blockIdx.x * wavesPerBlock + (threadIdx.x >> 5);
    const int tiles_n = N >> 5;              // 32-wide tiles
    const int tiles_m = (M + 15) >> 4;
    if (wave >= tiles_m * tiles_n) return;
    const int m0 = (wave / tiles_n) << 4;
    const int n0 = (wave % tiles_n) << 5;
    const int half = lane >> 4;              // 0: K,K+1   1: K+2,K+3
    const int l16  = lane & 15;

    v8f acc0 = {}, acc1 = {};
    int arow = m0 + l16;
    if (arow >= M) arow = 0;                 // clamp; result rows never stored
    const float* Arow = A + (size_t)arow * K + 2 * half;
    const float* Bp0  = B + (size_t)(2 * half) * N + n0 + l16;
    const float* Bp1  = Bp0 + 16;

    for (int k0 = 0; k0 < K; k0 += 4) {
        v2f a, b0, b1;
        a.x  = Arow[k0];
        a.y  = Arow[k0 + 1];
        b0.x = Bp0[(size_t)k0 * N];
        b0.y = Bp0[(size_t)k0 * N + N];
        b1.x = Bp1[(size_t)k0 * N];
        b1.y = Bp1[(size_t)k0 * N + N];
        acc0 = __builtin_amdgcn_wmma_f32_16x16x4_f32(
            false, a, false, b0, (short)0, acc0, false, false);
        acc1 = __builtin_amdgcn_wmma_f32_16x16x4_f32(
            false, a, false, b1, (short)0, acc1, false, false);
    }

    const int col0 = n0 + l16;
    const float bv0 = bias ? bias[col0] : 0.0f;
    const float bv1 = bias ? bias[col0 + 16] : 0.0f;
#pragma unroll
    for (int r = 0; r < 8; ++r) {
        int row = m0 + r + 8 * half;
        if (row < M) {
            float v0 = acc0[r] + bv0;
            float v1 = acc1[r] + bv1;
            if (do_relu) { v0 = fmaxf(v0, 0.0f); v1 = fmaxf(v1, 0.0f); }
            C[(size_t)row * N + col0]      = v0;
            C[(size_t)row * N + col0 + 16] = v1;
        }
    }
}

// --------------------------------------------- edge gather * w -> atomic scatter
// F/4 threads per edge, float4 per thread (m and agg are L2-resident: <=25.6MB).
template <int F>
__global__ __launch_bounds__(256)
void scatter_kernel(const int* __restrict__ src, const int* __restrict__ dst,
                    const float* __restrict__ nw,
                    const float* __restrict__ m, float* __restrict__ agg, int nE) {
    const int TPE = F / 4;
    long t = (long)blockIdx.x * blockDim.x + threadIdx.x;
    long e = t / TPE;
    if (e >= nE) return;
    int f4 = (int)(t % TPE) * 4;
    int s = src[e], d = dst[e];
    float w = nw[e];
    float4 mv = *(const float4*)(m + (size_t)s * F + f4);
    float* out = agg + (size_t)d * F + f4;
    atomicAdd(out + 0, mv.x * w);
    atomicAdd(out + 1, mv.y * w);
    atomicAdd(out + 2, mv.z * w);
    atomicAdd(out + 3, mv.w * w);
}

// h = relu(agg + dinv[i]^2 * m[i]  (self-loop) + b)   (written in place over agg)
template <int F>
__global__ void bias_relu_self_kernel(float* __restrict__ agg,
                                      const float* __restrict__ m,
                                      const float* __restrict__ dinv,
                                      const float* __restrict__ b, int n) {
    long idx = (long)blockIdx.x * blockDim.x + threadIdx.x;
    if (idx >= (long)n * F) return;
    int i = (int)(idx / F);
    int f = (int)(idx % F);
    float di = dinv[i];
    float v = agg[idx] + di * di * m[idx] + b[f];
    agg[idx] = fmaxf(v, 0.0f);
}

// g[graph[i], :] += h1[i, :]   (8 threads/node, float4 each; g fits in L2)
__global__ __launch_bounds__(256)
void pool_kernel(const float* __restrict__ h, const int* __restrict__ gidx,
                 float* __restrict__ g, int n) {
    long t = (long)blockIdx.x * blockDim.x + threadIdx.x;
    long i = t / 8;
    if (i >= n) return;
    int f4 = (int)(t % 8) * 4;
    float4 hv = *(const float4*)(h + (size_t)i * 32 + f4);
    float* out = g + (size_t)gidx[i] * 32 + f4;
    atomicAdd(out + 0, hv.x);
    atomicAdd(out + 1, hv.y);
    atomicAdd(out + 2, hv.z);
    atomicAdd(out + 3, hv.w);
}

// logits = gh @ Wm2 + bm2   (1000 x 128 @ 128 x 2)
__global__ void head_kernel(const float* __restrict__ gh,
                            const float* __restrict__ Wm2,
                            const float* __restrict__ bm2,
                            float* __restrict__ out, int n) {
    int i = blockIdx.x * blockDim.x + threadIdx.x;
    if (i >= n) return;
    float a0 = bm2[0], a1 = bm2[1];
    const float* row = gh + (size_t)i * 128;
#pragma unroll 4
    for (int k = 0; k < 128; ++k) {
        float v = row[k];
        a0 += v * Wm2[k * 2 + 0];
        a1 += v * Wm2[k * 2 + 1];
    }
    out[i * 2 + 0] = a0;
    out[i * 2 + 1] = a1;
}

// ---------------------------------------------------------------------- launch
extern "C" void kernel_launch(void* const* d_in, const int* in_sizes, int n_in,
                              void* d_out, int out_size, void* d_ws, size_t ws_size,
                              hipStream_t stream) {
    const float* x    = (const float*)d_in[0];
    const int*   ei   = (const int*)d_in[1];     // [2, E]
    const float* ew   = (const float*)d_in[2];
    const int*   gidx = (const int*)d_in[3];
    const float* W0   = (const float*)d_in[4];
    const float* b0   = (const float*)d_in[5];
    const float* W1   = (const float*)d_in[6];
    const float* b1   = (const float*)d_in[7];
    const float* Wm1  = (const float*)d_in[8];
    const float* bm1  = (const float*)d_in[9];
    const float* Wm2  = (const float*)d_in[10];
    const float* bm2  = (const float*)d_in[11];
    float* out = (float*)d_out;

    const int* src = ei;
    const int* dst = ei + N_EDGES;

    // workspace layout (floats)
    float* ws   = (float*)d_ws;
    float* dinv = ws;                              // N
    float* nw   = dinv + N_NODES;                  // E
    float* m0   = nw + N_EDGES;                    // N*64
    float* h0   = m0 + (size_t)N_NODES * 64;       // N*64 (agg0 -> h0)
    float* m1   = h0 + (size_t)N_NODES * 64;       // N*32
    float* h1   = m1 + (size_t)N_NODES * 32;       // N*32 (agg1 -> h1)
    float* g    = h1 + (size_t)N_NODES * 32;       // N_GRAPHS*32
    float* gh   = g + (size_t)N_GRAPHS * 32;       // N_GRAPHS*128

    const int B = 256;
    auto blocks = [](long n, int b) { return (int)((n + b - 1) / b); };

    // ---- GCN normalization
    fill_kernel<<<blocks(N_NODES, B), B, 0, stream>>>(dinv, 1.0f, N_NODES); // self-loop
    deg_kernel<<<blocks(N_EDGES, B), B, 0, stream>>>(dst, ew, dinv, N_EDGES);
    dinv_kernel<<<blocks(N_NODES, B), B, 0, stream>>>(dinv, N_NODES);
    normw_kernel<<<blocks(N_EDGES, B), B, 0, stream>>>(src, dst, ew, dinv, nw, N_EDGES);

    // ---- layer 0: m0 = x @ W0 (100000x128 @ 128x64)
    {
        long waves = (long)(N_NODES / 16) * (64 / 32);
        gemm16_wmma<<<blocks(waves, 8), 256, 0, stream>>>(x, W0, nullptr, m0,
                                                          N_NODES, 64, F_IN, 0);
    }
    fill_kernel<<<blocks((long)N_NODES * 64, B), B, 0, stream>>>(h0, 0.0f, (long)N_NODES * 64);
    scatter_kernel<64><<<blocks((long)N_EDGES * 16, B), B, 0, stream>>>(src, dst, nw, m0, h0, N_EDGES);
    bias_relu_self_kernel<64><<<blocks((long)N_NODES * 64, B), B, 0, stream>>>(h0, m0, dinv, b0, N_NODES);

    // ---- layer 1: m1 = h0 @ W1 (100000x64 @ 64x32)
    {
        long waves = (long)(N_NODES / 16) * (32 / 32);
        gemm16_wmma<<<blocks(waves, 8), 256, 0, stream>>>(h0, W1, nullptr, m1,
                                                          N_NODES, 32, 64, 0);
    }
    fill_kernel<<<blocks((long)N_NODES * 32, B), B, 0, stream>>>(h1, 0.0f, (long)N_NODES * 32);
    scatter_kernel<32><<<blocks((long)N_EDGES * 8, B), B, 0, stream>>>(src, dst, nw, m1, h1, N_EDGES);
    bias_relu_self_kernel<32><<<blocks((long)N_NODES * 32, B), B, 0, stream>>>(h1, m1, dinv, b1, N_NODES);

    // ---- graph sum pool
    fill_kernel<<<blocks((long)N_GRAPHS * 32, B), B, 0, stream>>>(g, 0.0f, (long)N_GRAPHS * 32);
    pool_kernel<<<blocks((long)N_NODES * 8, B), B, 0, stream>>>(h1, gidx, g, N_NODES);

    // ---- head: gh = relu(g @ Wm1 + bm1) (1000x32 @ 32x128), logits = gh @ Wm2 + bm2
    {
        long waves = (long)((N_GRAPHS + 15) / 16) * (128 / 32);
        gemm16_wmma<<<blocks(waves, 8), 256, 0, stream>>>(g, Wm1, bm1, gh,
                                                          N_GRAPHS, 128, 32, 1);
    }
    head_kernel<<<blocks(N_GRAPHS, B), B, 0, stream>>>(gh, Wm2, bm2, out, N_GRAPHS);
}